// FasterRCNNmps_43198781063712
// MI455X (gfx1250) — compile-verified
//
#include <hip/hip_runtime.h>
#include <math.h>

#define NA 250000
#define NG 64
#define NP 4096
#define LOWT 0.3f
#define HIGHT 0.7f
#define SBETA (1.0f / 9.0f)
#define SLICES 64

typedef __attribute__((ext_vector_type(2))) float v2f;
typedef __attribute__((ext_vector_type(8))) float v8f;

// Shared IoU routine: MUST be the single definition used by both passes so the
// float equality test (iou == best_per_gt) is bit-consistent.
__device__ __forceinline__ float iou_pair(float4 g, float areaG, float4 a, float areaA) {
    float ltx = fmaxf(g.x, a.x);
    float lty = fmaxf(g.y, a.y);
    float rbx = fminf(g.z, a.z);
    float rby = fminf(g.w, a.w);
    float w = fmaxf(rbx - ltx, 0.0f);
    float h = fmaxf(rby - lty, 0.0f);
    float inter = w * h;
    float uni = areaA + areaG - inter;
    return inter / uni;
}

__device__ __forceinline__ float smooth_l1(float d) {
    float ad = fabsf(d);
    return ad < SBETA ? 0.5f * d * d / SBETA : ad - 0.5f * SBETA;
}

// Sum 256 floats living in LDS using the matrix pipe.
// Four accumulating V_WMMA_F32_16X16X4_F32 with an all-ones B: each WMMA adds
// the row-sums of a 16x4 chunk (2 values per lane) into the f32 accumulator.
// After 4 WMMAs, VGPR j of D holds (replicated over N) the row sum for
// m = j (lanes 0-15) and m = j+8 (lanes 16-31). Per-lane add of the 8 D VGPRs
// + a lane^16 shuffle-add yields the grand total in every lane.
// Caller must be a fully-active wave (EXEC all ones).
__device__ __forceinline__ float wave_wmma_sum256(const float* buf, int lane) {
    v2f ones;
    ones[0] = 1.0f;
    ones[1] = 1.0f;
    v8f c = {};
#pragma unroll
    for (int i = 0; i < 4; ++i) {
        v2f a;
        a[0] = buf[lane * 8 + 2 * i + 0];
        a[1] = buf[lane * 8 + 2 * i + 1];
        c = __builtin_amdgcn_wmma_f32_16x16x4_f32(false, a, false, ones,
                                                  (short)0, c, false, false);
    }
    float s = c[0] + c[1] + c[2] + c[3] + c[4] + c[5] + c[6] + c[7];
    s += __shfl_xor(s, 16, 32);
    return s;
}

// ---------------- kernel 0: init accumulators -------------------------------
// ws[0..63]  : best_per_gt (as float bits; 0.0f == uint 0, valid for atomicMax
//              because all IoUs are >= 0)
// ws[64]     : n_samp   ws[65]: rpn box numerator   ws[66]: bce numerator
// ws[67]     : classification sum                   ws[68]: roi box sum
__global__ void k_init(float* ws) {
    int t = threadIdx.x;
    if (t < NG + 8) ws[t] = 0.0f;
}

// ---------------- kernel 1: best IoU per gt box -----------------------------
__global__ void k_best_per_gt(const float4* __restrict__ gt,
                              const float4* __restrict__ anchors,
                              unsigned int* __restrict__ best_u) {
    __shared__ float smax[256];
    int g = blockIdx.x / SLICES;
    int slice = blockIdx.x % SLICES;
    float4 gb = gt[g];
    float areaG = (gb.z - gb.x) * (gb.w - gb.y);
    const int chunk = (NA + SLICES - 1) / SLICES;
    int base = slice * chunk;
    int end = min(base + chunk, NA);
    float m = 0.0f;
    for (int a = base + (int)threadIdx.x; a < end; a += blockDim.x) {
        float4 ab = anchors[a];
        float areaA = (ab.z - ab.x) * (ab.w - ab.y);
        m = fmaxf(m, iou_pair(gb, areaG, ab, areaA));
    }
    smax[threadIdx.x] = m;
    __syncthreads();
    for (int s = 128; s > 0; s >>= 1) {
        if ((int)threadIdx.x < s)
            smax[threadIdx.x] = fmaxf(smax[threadIdx.x], smax[threadIdx.x + s]);
        __syncthreads();
    }
    if (threadIdx.x == 0) atomicMax(&best_u[g], __float_as_uint(smax[0]));
}

// ---------------- kernel 2: fused matcher + RPN losses ----------------------
__global__ void k_rpn(const float4* __restrict__ anchors,
                      const float4* __restrict__ gt,
                      const float* __restrict__ gt_score,
                      const int* __restrict__ gt_conf,
                      const float* __restrict__ objn,
                      const float4* __restrict__ deltas,
                      const float* __restrict__ best_f,
                      float* __restrict__ acc) {  // acc[0]=cnt acc[1]=box acc[2]=bce
    __shared__ float4 s_gt[NG];
    __shared__ float s_area[NG];
    __shared__ float s_score[NG];
    __shared__ float s_best[NG];
    __shared__ int s_conf[NG];
    __shared__ float s_red[3 * 256];

    int tid = threadIdx.x;
    if (tid < NG) {
        float4 gb = gt[tid];
        s_gt[tid] = gb;
        s_area[tid] = (gb.z - gb.x) * (gb.w - gb.y);
        s_score[tid] = gt_score[tid];
        s_conf[tid] = gt_conf[tid];
        s_best[tid] = best_f[tid];
    }
    __syncthreads();

    int a = blockIdx.x * blockDim.x + tid;
    float p_cnt = 0.0f, p_box = 0.0f, p_bce = 0.0f;
    if (a < NA) {
        float4 ab = anchors[a];
        float areaA = (ab.z - ab.x) * (ab.w - ab.y);
        float bestv = -1.0f;
        int besti = 0;
        bool force = false;
#pragma unroll 4
        for (int g = 0; g < NG; ++g) {
            float v = iou_pair(s_gt[g], s_area[g], ab, areaA);
            if (v > bestv) { bestv = v; besti = g; }  // first-max ties like argmax
            force = force || (v == s_best[g]);
        }
        int mi;
        if (force) mi = besti;
        else if (bestv < LOWT) mi = -1;
        else if (bestv < HIGHT) mi = -2;
        else mi = besti;

        int cl = mi >= 0 ? mi : 0;
        float lab;
        if (mi >= 0) {
            lab = fminf(1.0f, s_score[cl]);
            if (s_conf[cl] == 0) lab = -1.0f;
        } else if (mi == -1) {
            lab = 0.0f;
        } else {
            lab = -1.0f;
        }

        bool pos = lab >= 1.0f;
        bool neg = lab == 0.0f;
        p_cnt = (pos ? 1.0f : 0.0f) + (neg ? 1.0f : 0.0f);

        if (pos) {
            float4 gb = s_gt[cl];
            float aw = ab.z - ab.x, ah = ab.w - ab.y;
            float acx = ab.x + 0.5f * aw, acy = ab.y + 0.5f * ah;
            float gw = gb.z - gb.x, gh = gb.w - gb.y;
            float gcx = gb.x + 0.5f * gw, gcy = gb.y + 0.5f * gh;
            float t0 = (gcx - acx) / aw;
            float t1 = (gcy - acy) / ah;
            float t2 = logf(gw / aw);
            float t3 = logf(gh / ah);
            float4 d = deltas[a];
            p_box = smooth_l1(d.x - t0) + smooth_l1(d.y - t1) +
                    smooth_l1(d.z - t2) + smooth_l1(d.w - t3);
        }
        if (pos || neg) {
            float y = fminf(fmaxf(lab, 0.0f), 1.0f);
            float x = objn[a];
            p_bce = fmaxf(x, 0.0f) - x * y + log1pf(expf(-fabsf(x)));
        }
    }
    s_red[0 * 256 + tid] = p_cnt;
    s_red[1 * 256 + tid] = p_box;
    s_red[2 * 256 + tid] = p_bce;
    __syncthreads();
    if (tid < 32) {
        for (int k = 0; k < 3; ++k) {
            float tot = wave_wmma_sum256(&s_red[k * 256], tid);
            if (tid == 0) atomicAdd(&acc[k], tot);
        }
    }
}

// ---------------- kernel 3: ROI classification + box losses -----------------
__global__ void k_roi(const float* __restrict__ logits,  // [P,2]
                      const float* __restrict__ boxreg,  // [P,8]
                      const int* __restrict__ labels,
                      const float4* __restrict__ tgts,   // [P,4]
                      const float* __restrict__ scores,
                      float* __restrict__ acc) {  // acc[0]=cls acc[1]=box
    __shared__ float s_red[2 * 256];
    int tid = threadIdx.x;
    int p = blockIdx.x * blockDim.x + tid;
    float p_cls = 0.0f, p_box = 0.0f;
    if (p < NP) {
        float l0 = logits[2 * p + 0];
        float l1 = logits[2 * p + 1];
        float m = fmaxf(l0, l1);
        float lse = m + logf(expf(l0 - m) + expf(l1 - m));
        float s = scores[p];
        p_cls = -((1.0f - s) * (l0 - lse) + s * (l1 - lse));

        int lb = labels[p];
        int cl = lb > 0 ? lb : 0;
        const float4* row = (const float4*)(boxreg + 8 * p);
        float4 b = row[cl];
        float4 t = tgts[p];
        float sl = smooth_l1(b.x - t.x) + smooth_l1(b.y - t.y) +
                   smooth_l1(b.z - t.z) + smooth_l1(b.w - t.w);
        p_box = (lb > 0) ? sl : 0.0f;
    }
    s_red[tid] = p_cls;
    s_red[256 + tid] = p_box;
    __syncthreads();
    if (tid < 32) {
        for (int k = 0; k < 2; ++k) {
            float tot = wave_wmma_sum256(&s_red[k * 256], tid);
            if (tid == 0) atomicAdd(&acc[k], tot);
        }
    }
}

// ---------------- kernel 4: finalize ----------------------------------------
__global__ void k_final(const float* __restrict__ ws, float* __restrict__ out) {
    if (threadIdx.x == 0 && blockIdx.x == 0) {
        float n = ws[64];
        out[0] = ws[66] / n;           // objectness (BCE) loss
        out[1] = ws[65] / n;           // rpn box loss
        out[2] = ws[67] / (float)NP;   // classification loss (mean)
        out[3] = ws[68] / (float)NP;   // roi box loss
    }
}

extern "C" void kernel_launch(void* const* d_in, const int* in_sizes, int n_in,
                              void* d_out, int out_size, void* d_ws, size_t ws_size,
                              hipStream_t stream) {
    const float4* anchors = (const float4*)d_in[0];
    const float4* gt      = (const float4*)d_in[1];
    const float* gt_score = (const float*)d_in[2];
    const int* gt_conf    = (const int*)d_in[3];
    const float* objn     = (const float*)d_in[4];
    const float4* deltas  = (const float4*)d_in[5];
    const float* logits   = (const float*)d_in[6];
    const float* boxreg   = (const float*)d_in[7];
    const int* rlabels    = (const int*)d_in[8];
    const float4* rtgts   = (const float4*)d_in[9];
    const float* rscores  = (const float*)d_in[10];
    float* ws = (float*)d_ws;
    float* out = (float*)d_out;

    k_init<<<1, 128, 0, stream>>>(ws);
    k_best_per_gt<<<NG * SLICES, 256, 0, stream>>>(gt, anchors, (unsigned int*)ws);
    k_rpn<<<(NA + 255) / 256, 256, 0, stream>>>(anchors, gt, gt_score, gt_conf,
                                                objn, deltas, ws, ws + 64);
    k_roi<<<(NP + 255) / 256, 256, 0, stream>>>(logits, boxreg, rlabels, rtgts,
                                                rscores, ws + 67);
    k_final<<<1, 1, 0, stream>>>(ws, out);
}